// NasAutoGraphA_36816459661699
// MI455X (gfx1250) — compile-verified
//
#include <hip/hip_runtime.h>
#include <hip/hip_bf16.h>

typedef float v2f __attribute__((ext_vector_type(2)));
typedef float v8f __attribute__((ext_vector_type(8)));

#define HID   64
#define HEADS 2
#define GSLOPE 0.2f

__device__ __forceinline__ unsigned f2o(float f) {
    unsigned u = __float_as_uint(f);
    return (u & 0x80000000u) ? ~u : (u | 0x80000000u);
}
__device__ __forceinline__ float o2f(unsigned u) {
    return __uint_as_float((u & 0x80000000u) ? (u & 0x7fffffffu) : ~u);
}
__device__ __forceinline__ float lrelu(float v, float s) { return v > 0.f ? v : s * v; }

// edge id -> (src, dst, weight), with virtual self-loops appended at e >= E
__device__ __forceinline__ void edge_sd(int e, const int* __restrict__ ei,
                                        const float* __restrict__ ew, int E, int N,
                                        int& s, int& d, float& w) {
    if (e < E) { s = ei[e]; d = ei[E + e]; w = ew[e]; }
    else       { s = d = e - E; w = 1.0f; }
}

// ---------------------------------------------------------------------------
// fp32 WMMA GEMM: C[M,Nout] = A[M,K] @ B[K,Nout] (+bias)
// 4 waves/block, each wave owns one 16x16 tile (4 stacked M-tiles, shared
// 16-col B tile staged in LDS in WMMA fragment-pair layout so the inner loop
// is: global_load_b64 (A frag) + ds_load_b64 (B frag) + v_wmma_f32_16x16x4.
// K%4==0, K<=256, Nout%16==0 at every call site; M may be ragged.
// ---------------------------------------------------------------------------
__global__ __launch_bounds__(128) void gemm_f32_wmma(
    const float* __restrict__ A, const float* __restrict__ B,
    const float* __restrict__ bias, float* __restrict__ C,
    int M, int K, int Nout)
{
    __shared__ float bs[256 * 16];              // (K/2)*16 v2f pairs, max 16 KB

    const int tid  = threadIdx.x;
    const int lane = tid & 31;
    const int wave = tid >> 5;
    const int lm   = lane & 15;
    const int half = lane >> 4;
    const int col0 = blockIdx.y * 16;

    // Cooperative stage of B[K,16] tile into pair layout:
    // pair kk = k/2 holds {B[2kk][n], B[2kk+1][n]} at float index kk*32 + 2n + (k&1)
    for (int idx = tid; idx < K * 16; idx += 128) {
        const int k = idx >> 4, n = idx & 15;
        bs[((k >> 1) * 16 + n) * 2 + (k & 1)] = B[(size_t)k * Nout + col0 + n];
    }
    __syncthreads();

    const int row0 = blockIdx.x * 64 + wave * 16;
    const int ra   = min(row0 + lm, M - 1);     // clamp loads; stores guarded below
    const float* __restrict__ Ap = A + (size_t)ra * K + 2 * half;
    const v2f*   __restrict__ Bp = (const v2f*)bs + half * 16 + lm;

    v8f acc = {};
#pragma unroll 4
    for (int k0 = 0; k0 < K; k0 += 4) {
        const v2f a = *(const v2f*)(Ap + k0);               // global_load_b64
        const v2f b = Bp[(k0 >> 1) * 16];                   // ds_load_b64
        acc = __builtin_amdgcn_wmma_f32_16x16x4_f32(
            /*neg_a=*/false, a, /*neg_b=*/false, b,
            /*c_mod=*/(short)0, acc, /*reuse_a=*/false, /*reuse_b=*/false);
    }

    const float bv = bias ? bias[col0 + lm] : 0.f;
#pragma unroll
    for (int r = 0; r < 8; ++r) {
        const int m = row0 + r + 8 * half;
        if (m < M) C[(size_t)m * Nout + col0 + lm] = acc[r] + bv;
    }
}

// ---------------------------------------------------------------------------
// Graph normalization (shared by both layers: degrees are layer-invariant)
// ---------------------------------------------------------------------------
__global__ void deg_kernel(const int* __restrict__ ei, const float* __restrict__ ew,
                           int E, int N, float* __restrict__ deg) {
    int e = blockIdx.x * blockDim.x + threadIdx.x;
    if (e >= E + N) return;
    int s, d; float w;
    edge_sd(e, ei, ew, E, N, s, d, w);
    atomicAdd(&deg[d], w);
}

__global__ void dinv_kernel(float* __restrict__ deg, int N) {
    int n = blockIdx.x * blockDim.x + threadIdx.x;
    if (n >= N) return;
    float v = deg[n];
    deg[n] = v > 0.f ? __frsqrt_rn(v) : 0.f;    // deg now holds d^-1/2
}

__global__ void norm_kernel(const int* __restrict__ ei, const float* __restrict__ ew,
                            const float* __restrict__ dinv, int E, int N,
                            float* __restrict__ enorm) {
    int e = blockIdx.x * blockDim.x + threadIdx.x;
    if (e >= E + N) return;
    int s, d; float w;
    edge_sd(e, ei, ew, E, N, s, d, w);
    enorm[e] = dinv[s] * w * dinv[d];
}

// ---------------------------------------------------------------------------
// GAT attention
// ---------------------------------------------------------------------------
__global__ void attn_kernel(const float* __restrict__ xp,
                            const float* __restrict__ att_src,
                            const float* __restrict__ att_dst,
                            float* __restrict__ asrc, float* __restrict__ adst, int N) {
    int idx = blockIdx.x * blockDim.x + threadIdx.x;
    if (idx >= N * HEADS) return;
    int n = idx >> 1, h = idx & 1;
    const float* row = xp + (size_t)n * (HEADS * HID) + h * HID;
    float s = 0.f, d = 0.f;
#pragma unroll 8
    for (int c = 0; c < HID; ++c) {
        float v = row[c];
        s += v * att_src[h * HID + c];
        d += v * att_dst[h * HID + c];
    }
    asrc[idx] = s;
    adst[idx] = d;
}

// segment max over dst via monotonic-uint atomicMax (memset-0 == -inf)
__global__ void edge_max_kernel(const int* __restrict__ ei, const float* __restrict__ ew,
                                const float* __restrict__ asrc, const float* __restrict__ adst,
                                unsigned* __restrict__ mbuf, int E, int N) {
    int e = blockIdx.x * blockDim.x + threadIdx.x;
    if (e >= E + N) return;
    int s, d; float w;
    edge_sd(e, ei, ew, E, N, s, d, w);
#pragma unroll
    for (int h = 0; h < HEADS; ++h) {
        float v = lrelu(asrc[s * HEADS + h] + adst[d * HEADS + h], GSLOPE);
        atomicMax(&mbuf[d * HEADS + h], f2o(v));
    }
}

__global__ void edge_exp_kernel(const int* __restrict__ ei, const float* __restrict__ ew,
                                const float* __restrict__ asrc, const float* __restrict__ adst,
                                const unsigned* __restrict__ mbuf, float* __restrict__ sbuf,
                                float* __restrict__ ebuf, int E, int N) {
    int e = blockIdx.x * blockDim.x + threadIdx.x;
    if (e >= E + N) return;
    int s, d; float w;
    edge_sd(e, ei, ew, E, N, s, d, w);
#pragma unroll
    for (int h = 0; h < HEADS; ++h) {
        float v = lrelu(asrc[s * HEADS + h] + adst[d * HEADS + h], GSLOPE);
        float m = o2f(mbuf[d * HEADS + h]);
        float t = __expf(v - m);
        ebuf[(size_t)e * HEADS + h] = t;
        atomicAdd(&sbuf[d * HEADS + h], t);
    }
}

// message scatter: out1[d, :] += xp[s, :] * alpha   (2 edges per 256-thread block)
__global__ __launch_bounds__(256) void edge_msg_kernel(
    const int* __restrict__ ei, const float* __restrict__ ew,
    const float* __restrict__ ebuf, const float* __restrict__ sbuf,
    const float* __restrict__ xp, float* __restrict__ out1, int E, int N) {
    int e = blockIdx.x * 2 + (threadIdx.x >> 7);
    if (e >= E + N) return;
    int c = threadIdx.x & 127;
    int h = c >> 6;
    int s, d; float w;
    edge_sd(e, ei, ew, E, N, s, d, w);
    float alpha = ebuf[(size_t)e * HEADS + h] / (sbuf[d * HEADS + h] + 1e-16f);
    atomicAdd(&out1[(size_t)d * (HEADS * HID) + c],
              xp[(size_t)s * (HEADS * HID) + c] * alpha);
}

// SGConv propagation: hagg[d, :] += xh[s, :] * enorm[e]   (4 edges per block)
__global__ __launch_bounds__(256) void sg_scatter_kernel(
    const int* __restrict__ ei, const float* __restrict__ ew,
    const float* __restrict__ enorm, const float* __restrict__ xh,
    float* __restrict__ hagg, int E, int N) {
    int e = blockIdx.x * 4 + (threadIdx.x >> 6);
    if (e >= E + N) return;
    int c = threadIdx.x & 63;
    int s, d; float w;
    edge_sd(e, ei, ew, E, N, s, d, w);
    atomicAdd(&hagg[(size_t)d * HID + c], xh[(size_t)s * HID + c] * enorm[e]);
}

// x2 = relu(concat(leaky(gat+b), leaky(sg)))  ; relu∘leaky == relu
__global__ void concat_relu_kernel(const float* __restrict__ out1,
                                   const float* __restrict__ gatb,
                                   const float* __restrict__ sgout,
                                   float* __restrict__ x2, int N) {
    size_t idx = (size_t)blockIdx.x * blockDim.x + threadIdx.x;
    size_t total = (size_t)N * 192;
    if (idx >= total) return;
    int n = (int)(idx / 192);
    int c = (int)(idx % 192);
    float v = (c < 128) ? out1[(size_t)n * 128 + c] + gatb[c]
                        : sgout[(size_t)n * 64 + (c - 128)];
    x2[idx] = fmaxf(v, 0.f);
}

// wave32 == 32 classes: one wave per node, shuffle reductions
__global__ __launch_bounds__(256) void log_softmax_kernel(
    const float* __restrict__ logits, float* __restrict__ out, int N, int NC) {
    int node = blockIdx.x * 8 + (threadIdx.x >> 5);
    int lane = threadIdx.x & 31;
    if (node >= N) return;
    float v = (lane < NC) ? logits[(size_t)node * NC + lane] : -INFINITY;
    float mx = v;
#pragma unroll
    for (int off = 16; off; off >>= 1) mx = fmaxf(mx, __shfl_xor(mx, off, 32));
    float ex = (lane < NC) ? __expf(v - mx) : 0.f;
    float sm = ex;
#pragma unroll
    for (int off = 16; off; off >>= 1) sm += __shfl_xor(sm, off, 32);
    if (lane < NC) out[(size_t)node * NC + lane] = v - mx - __logf(sm);
}

// ---------------------------------------------------------------------------
extern "C" void kernel_launch(void* const* d_in, const int* in_sizes, int n_in,
                              void* d_out, int out_size, void* d_ws, size_t ws_size,
                              hipStream_t stream) {
    (void)n_in; (void)out_size; (void)ws_size;

    const float* x  = (const float*)d_in[0];
    const int*   ei = (const int*)d_in[1];
    const float* ew = (const float*)d_in[2];

    const int HIDv = in_sizes[4];              // 64
    const int F_IN = in_sizes[3] / HIDv;       // 256
    const int N    = in_sizes[0] / F_IN;       // 100000
    const int E    = in_sizes[2];              // 1200000
    const int NC   = in_sizes[20];             // 32
    const int Et   = E + N;

    float* ws = (float*)d_ws;
    size_t off = 0;
    float*    XH    = ws + off; off += (size_t)N * 64;
    float*    XP    = ws + off; off += (size_t)N * 128;
    float*    OUT1  = ws + off; off += (size_t)N * 128;
    float*    HAGG  = ws + off; off += (size_t)N * 64;
    float*    SGOUT = ws + off; off += (size_t)N * 64;
    float*    X2    = ws + off; off += (size_t)N * 192;
    float*    ASRC  = ws + off; off += (size_t)N * 2;
    float*    ADST  = ws + off; off += (size_t)N * 2;
    unsigned* MBUF  = (unsigned*)(ws + off); off += (size_t)N * 2;
    float*    SBUF  = ws + off; off += (size_t)N * 2;
    float*    DEG   = ws + off; off += (size_t)N;
    float*    ENORM = ws + off; off += (size_t)Et;
    float*    EBUF  = ws + off; off += (size_t)Et * 2;
    float*    LOGITS = XH;                     // XH is dead by classifier time

    const int TB = 256;
    const int egrid = (Et + TB - 1) / TB;

    // gcn_norm once (degrees identical for both layers)
    hipMemsetAsync(DEG, 0, (size_t)N * sizeof(float), stream);
    deg_kernel<<<egrid, TB, 0, stream>>>(ei, ew, E, N, DEG);
    dinv_kernel<<<(N + TB - 1) / TB, TB, 0, stream>>>(DEG, N);
    norm_kernel<<<egrid, TB, 0, stream>>>(ei, ew, DEG, E, N, ENORM);

    const int mgrid = (N + 63) / 64;           // 4 waves/block, 16 rows/wave

    const float* XIN = x;
    int K = F_IN;
    for (int l = 0; l < 2; ++l) {
        const int pb = 3 + l * 8;
        const float* preW = (const float*)d_in[pb + 0];
        const float* preb = (const float*)d_in[pb + 1];
        const float* gatW = (const float*)d_in[pb + 2];
        const float* aS   = (const float*)d_in[pb + 3];
        const float* aD   = (const float*)d_in[pb + 4];
        const float* gatb = (const float*)d_in[pb + 5];
        const float* sgW  = (const float*)d_in[pb + 6];
        const float* sgb  = (const float*)d_in[pb + 7];

        // xh = x @ pre_W + pre_b
        gemm_f32_wmma<<<dim3(mgrid, 64 / 16), 128, 0, stream>>>(
            XIN, preW, preb, XH, N, K, 64);
        // xp = xh @ gat_W  (bias applied after aggregation)
        gemm_f32_wmma<<<dim3(mgrid, 128 / 16), 128, 0, stream>>>(
            XH, gatW, nullptr, XP, N, 64, 128);

        attn_kernel<<<(N * HEADS + TB - 1) / TB, TB, 0, stream>>>(XP, aS, aD, ASRC, ADST, N);

        hipMemsetAsync(MBUF, 0, (size_t)N * 2 * sizeof(unsigned), stream); // == -inf encoded
        hipMemsetAsync(SBUF, 0, (size_t)N * 2 * sizeof(float), stream);
        hipMemsetAsync(OUT1, 0, (size_t)N * 128 * sizeof(float), stream);
        hipMemsetAsync(HAGG, 0, (size_t)N * 64 * sizeof(float), stream);

        edge_max_kernel<<<egrid, TB, 0, stream>>>(ei, ew, ASRC, ADST, MBUF, E, N);
        edge_exp_kernel<<<egrid, TB, 0, stream>>>(ei, ew, ASRC, ADST, MBUF, SBUF, EBUF, E, N);
        edge_msg_kernel<<<(Et + 1) / 2, TB, 0, stream>>>(ei, ew, EBUF, SBUF, XP, OUT1, E, N);

        sg_scatter_kernel<<<(Et + 3) / 4, TB, 0, stream>>>(ei, ew, ENORM, XH, HAGG, E, N);
        gemm_f32_wmma<<<dim3(mgrid, 64 / 16), 128, 0, stream>>>(
            HAGG, sgW, sgb, SGOUT, N, 64, 64);

        concat_relu_kernel<<<(int)(((size_t)N * 192 + TB - 1) / TB), TB, 0, stream>>>(
            OUT1, gatb, SGOUT, X2, N);

        XIN = X2;
        K = 192;
    }

    const float* clsW = (const float*)d_in[19];
    const float* clsb = (const float*)d_in[20];
    gemm_f32_wmma<<<dim3(mgrid, NC / 16), 128, 0, stream>>>(
        X2, clsW, clsb, LOGITS, N, 192, NC);
    log_softmax_kernel<<<(N + 7) / 8, TB, 0, stream>>>(LOGITS, (float*)d_out, N, NC);
}